// VectorQuantizerEMA_31121333026983
// MI455X (gfx1250) — compile-verified
//
#include <hip/hip_runtime.h>
#include <math.h>

// ---------------------------------------------------------------------------
// VQ-VAE EMA vector quantizer for MI455X (gfx1250, wave32, WMMA).
// - distance GEMM via exact V_WMMA_F32_16X16X4_F32 (argmin needs exactness;
//   total compute is only 8.6 GFLOP, kernel is HBM-bound by the 256MB
//   one-hot write -> low-precision WMMA buys nothing).
// - one-hot / z_q outputs are streamed with non-temporal stores so the
//   256MB write stream does not thrash the 192MB L2 (codebook + z stay hot).
// ---------------------------------------------------------------------------

typedef float v2f __attribute__((ext_vector_type(2)));
typedef float v4f __attribute__((ext_vector_type(4)));
typedef float v8f __attribute__((ext_vector_type(8)));

#define N_CODES 1024
#define DIM     64
#define BATCH   65536
#define DECAY   0.99f
#define OMD     0.01f          // 1 - DECAY
#define EPS     1e-5f

// ---- output layout (floats, concatenated in reference return order) -------
#define ZQ_OFF    0
#define LOSS_OFF  (BATCH * DIM)               // 4,194,304
#define PERP_OFF  (LOSS_OFF + 1)
#define USAGE_OFF (LOSS_OFF + 2)
#define IDX_OFF   (LOSS_OFF + 3)
#define ENC_OFF   (IDX_OFF + BATCH)
#define NCB_OFF   (ENC_OFF + (size_t)BATCH * N_CODES)
#define NCS_OFF   (NCB_OFF + N_CODES * DIM)
#define NEA_OFF   (NCS_OFF + N_CODES)

// ---- workspace layout (floats) --------------------------------------------
#define WS_CNORM   0
#define WS_CLUSTER (WS_CNORM + N_CODES)
#define WS_EMB     (WS_CLUSTER + N_CODES)
#define WS_COMMIT  (WS_EMB + N_CODES * DIM)   // total ~270 KB

#define CB_STRIDE   68          // 64 + 4 pad floats -> conflict-free ds_load_b64
#define ROWS_PER_WG 128         // 8 waves x 16 rows

// ===========================================================================
// K0: zero scatter buffers, compute per-code squared norms ||c||^2
// ===========================================================================
__global__ void vq_init(const float* __restrict__ cb, float* __restrict__ ws) {
    int gt = blockIdx.x * blockDim.x + threadIdx.x;
    if (gt < N_CODES * DIM) ws[WS_EMB + gt] = 0.0f;
    if (gt < N_CODES) {
        ws[WS_CLUSTER + gt] = 0.0f;
        const float* row = cb + gt * DIM;
        float s = 0.0f;
#pragma unroll
        for (int d = 0; d < DIM; d += 4) {
            v4f v = *(const v4f*)(row + d);
            s += v.x * v.x + v.y * v.y + v.z * v.z + v.w * v.w;
        }
        ws[WS_CNORM + gt] = s;
    }
    if (gt == 0) ws[WS_COMMIT] = 0.0f;
}

// ===========================================================================
// K1: fused distance-GEMM (f32 WMMA) + argmin + one-hot + z_q + scatters
// ===========================================================================
__global__ __launch_bounds__(256) void vq_main(const float* __restrict__ z,
                                               const float* __restrict__ cb,
                                               float* __restrict__ out,
                                               float* __restrict__ ws) {
    extern __shared__ float smem[];
    float* cb_lds    = smem;                          // 1024 * 68 floats (padded)
    float* cnorm_lds = smem + N_CODES * CB_STRIDE;    // 1024 floats

    const int tid  = threadIdx.x;
    const int wave = tid >> 5;
    const int lane = tid & 31;
    const int hl   = lane >> 4;       // lane half (A/B K-split per ISA layout)
    const int l15  = lane & 15;

    // ---- stage codebook into padded LDS (coalesced b128 loads) ------------
    for (int i = tid; i < N_CODES * (DIM / 4); i += 256) {
        int row = i >> 4;
        int c4  = (i & 15) << 2;
        *(v4f*)(cb_lds + row * CB_STRIDE + c4) = *(const v4f*)(cb + row * DIM + c4);
    }
    for (int i = tid; i < N_CODES; i += 256) cnorm_lds[i] = ws[WS_CNORM + i];
    __syncthreads();

    const int r0 = blockIdx.x * ROWS_PER_WG + wave * 16;

    // ---- A fragments: entire 16x64 z strip in registers (32 VGPRs) --------
    // f32 A 16x4 layout: lanes 0-15 hold K={0,1}, lanes 16-31 hold K={2,3}.
    v2f afrag[16];
    {
        const float* zr = z + (size_t)(r0 + l15) * DIM + 2 * hl;
#pragma unroll
        for (int k = 0; k < 16; ++k) afrag[k] = *(const v2f*)(zr + 4 * k);
    }

    // ---- sweep 1024 codes, two 16-code tiles in flight ---------------------
    float best[8];
    int   bestn[8];
#pragma unroll
    for (int v = 0; v < 8; ++v) { best[v] = 3.4e38f; bestn[v] = 0; }

    for (int g = 0; g < 64; g += 2) {
        v8f acc0 = {0.f, 0.f, 0.f, 0.f, 0.f, 0.f, 0.f, 0.f};
        v8f acc1 = acc0;
        const float* bp0 = cb_lds + (g * 16 + l15) * CB_STRIDE + 2 * hl;
        const float* bp1 = bp0 + 16 * CB_STRIDE;
#pragma unroll
        for (int k = 0; k < 16; ++k) {
            v2f f0 = *(const v2f*)(bp0 + 4 * k);    // ds_load_2addr_b64, conflict-free
            v2f f1 = *(const v2f*)(bp1 + 4 * k);
            acc0 = __builtin_amdgcn_wmma_f32_16x16x4_f32(
                false, afrag[k], false, f0, (short)0, acc0, false, false);
            acc1 = __builtin_amdgcn_wmma_f32_16x16x4_f32(
                false, afrag[k], false, f1, (short)0, acc1, false, false);
        }
        // score = ||c||^2 - 2 z.c   (||z||^2 constant per row; drop for argmin)
        float cn0 = cnorm_lds[g * 16 + l15];
        float cn1 = cnorm_lds[g * 16 + 16 + l15];
        int   n0  = g * 16 + l15;
        int   n1  = n0 + 16;
#pragma unroll
        for (int v = 0; v < 8; ++v) {
            float s0 = __builtin_fmaf(-2.0f, acc0[v], cn0);
            if (s0 < best[v]) { best[v] = s0; bestn[v] = n0; }
            float s1 = __builtin_fmaf(-2.0f, acc1[v], cn1);
            if (s1 < best[v]) { best[v] = s1; bestn[v] = n1; }
        }
    }

    // ---- argmin reduce across the 16 lanes of each half (rows stay put) ---
#pragma unroll
    for (int off = 1; off < 16; off <<= 1) {
#pragma unroll
        for (int v = 0; v < 8; ++v) {
            float ob = __shfl_xor(best[v], off, 32);
            int   on = __shfl_xor(bestn[v], off, 32);
            if (ob < best[v] || (ob == best[v] && on < bestn[v])) {
                best[v] = ob; bestn[v] = on;
            }
        }
    }
    // D layout: VGPR v: lanes 0-15 -> row v ; lanes 16-31 -> row v+8
    int rowidx[16];
#pragma unroll
    for (int m = 0; m < 16; ++m)
        rowidx[m] = __shfl(bestn[m & 7], (m < 8) ? 0 : 16, 32);

    // ---- per-row epilogue: one-hot, z_q, commitment, scatter-sums ---------
    float commit = 0.0f;
#pragma unroll 1
    for (int m = 0; m < 16; ++m) {
        const int    r   = r0 + m;
        const int    idx = rowidx[m];
        // one-hot row: 256MB total stream, write-once -> non-temporal b128
        float* enc = out + ENC_OFF + (size_t)r * N_CODES;
#pragma unroll
        for (int j = lane; j < 256; j += 32) {
            v4f val;
            val.x = (idx == 4 * j + 0) ? 1.0f : 0.0f;
            val.y = (idx == 4 * j + 1) ? 1.0f : 0.0f;
            val.z = (idx == 4 * j + 2) ? 1.0f : 0.0f;
            val.w = (idx == 4 * j + 3) ? 1.0f : 0.0f;
            __builtin_nontemporal_store(val, (v4f*)(enc + 4 * j));
        }
        const int d = lane * 2;
        v2f q  = *(const v2f*)(cb_lds + idx * CB_STRIDE + d);
        v2f zv = *(const v2f*)(z + (size_t)r * DIM + d);
        __builtin_nontemporal_store(q, (v2f*)(out + ZQ_OFF + (size_t)r * DIM + d));
        float dx = zv.x - q.x, dy = zv.y - q.y;
        commit += dx * dx + dy * dy;
        atomicAdd(ws + WS_EMB + idx * DIM + d,     zv.x);
        atomicAdd(ws + WS_EMB + idx * DIM + d + 1, zv.y);
        if (lane == 0) {
            out[IDX_OFF + r] = (float)idx;
            atomicAdd(ws + WS_CLUSTER + idx, 1.0f);
        }
    }
#pragma unroll
    for (int off = 16; off >= 1; off >>= 1) commit += __shfl_xor(commit, off, 32);
    if (lane == 0) atomicAdd(ws + WS_COMMIT, commit);
}

// ===========================================================================
// K2: EMA updates, normalization, perplexity/usage/loss scalars
// ===========================================================================
__global__ __launch_bounds__(1024) void vq_final(const float* __restrict__ cluster_size,
                                                 const float* __restrict__ embed_avg,
                                                 float* __restrict__ out,
                                                 const float* __restrict__ ws) {
    __shared__ float rN[1024];
    __shared__ float rP[1024];
    __shared__ float rU[1024];
    const int t = threadIdx.x;

    float bc  = ws[WS_CLUSTER + t];
    float ncs = cluster_size[t] * DECAY + OMD * bc;
    out[NCS_OFF + t] = ncs;

    float p = bc * (1.0f / BATCH);
    rN[t] = ncs;
    rP[t] = p * logf(p + 1e-10f);
    rU[t] = (p > 1e-3f) ? 1.0f : 0.0f;
    __syncthreads();
    for (int s = 512; s > 0; s >>= 1) {
        if (t < s) { rN[t] += rN[t + s]; rP[t] += rP[t + s]; rU[t] += rU[t + s]; }
        __syncthreads();
    }
    float ntot   = rN[0];
    float cs     = (ncs + EPS) / (ntot + N_CODES * EPS) * ntot;
    float inv_cs = 1.0f / cs;
#pragma unroll
    for (int d = 0; d < DIM; d += 4) {
        v4f es  = *(const v4f*)(ws + WS_EMB + t * DIM + d);
        v4f ea  = *(const v4f*)(embed_avg + t * DIM + d);
        v4f nea = ea * DECAY + es * OMD;
        *(v4f*)(out + NEA_OFF + t * DIM + d) = nea;
        *(v4f*)(out + NCB_OFF + t * DIM + d) = nea * inv_cs;
    }
    if (t == 0) {
        out[LOSS_OFF]  = ws[WS_COMMIT] * (1.0f / ((float)BATCH * DIM));
        out[PERP_OFF]  = expf(-rP[0]);
        out[USAGE_OFF] = rU[0] * (1.0f / N_CODES);
    }
}

// ===========================================================================
extern "C" void kernel_launch(void* const* d_in, const int* in_sizes, int n_in,
                              void* d_out, int out_size, void* d_ws, size_t ws_size,
                              hipStream_t stream) {
    const float* z            = (const float*)d_in[0];
    const float* cb           = (const float*)d_in[1];
    const float* cluster_size = (const float*)d_in[2];
    const float* embed_avg    = (const float*)d_in[3];
    float* out = (float*)d_out;
    float* ws  = (float*)d_ws;

    vq_init<<<(N_CODES * DIM + 255) / 256, 256, 0, stream>>>(cb, ws);

    size_t lds_bytes = (size_t)(N_CODES * CB_STRIDE + N_CODES) * sizeof(float); // ~283 KB
    vq_main<<<BATCH / ROWS_PER_WG, 256, lds_bytes, stream>>>(z, cb, out, ws);

    vq_final<<<1, 1024, 0, stream>>>(cluster_size, embed_avg, out, ws);
}